// GCNNonlinearRegressionModel_66907000537715
// MI455X (gfx1250) — compile-verified
//
#include <hip/hip_runtime.h>
#include <hip/hip_bf16.h>

#define N_NODES   50000
#define N_EDGES   800000
#define NUM_GRAPHS 64
#define DH        128          // D_IN == D_HID == D_EMB == 128
#define D_PRED    256
#define BN_EPS    1e-5f

#define KPAIRS    (DH / 2)     // 64 K-pairs
#define PSTRIDE   288          // dwords per kp-row in LDS (256 data + 32 pad)

typedef __attribute__((ext_vector_type(2))) float v2f;
typedef __attribute__((ext_vector_type(8))) float v8f;

// ---------------------------------------------------------------------------
// Generic zero-fill
// ---------------------------------------------------------------------------
__global__ void zero_kernel(float* __restrict__ p, size_t n) {
    size_t i = (size_t)blockIdx.x * blockDim.x + threadIdx.x;
    if (i < n) p[i] = 0.0f;
}

// ---------------------------------------------------------------------------
// Degree: deg[dst[e]] += 1
// ---------------------------------------------------------------------------
__global__ void degree_kernel(const int* __restrict__ dst, float* __restrict__ deg,
                              int nEdges) {
    int e = blockIdx.x * blockDim.x + threadIdx.x;
    if (e < nEdges) atomicAdd(&deg[dst[e]], 1.0f);
}

// invs[i] = 1/sqrt(deg[i]+1)   (self-loop included)
__global__ void invsqrt_kernel(const float* __restrict__ deg, float* __restrict__ invs,
                               int n) {
    int i = blockIdx.x * blockDim.x + threadIdx.x;
    if (i < n) invs[i] = rsqrtf(deg[i] + 1.0f);
}

// ---------------------------------------------------------------------------
// WMMA f32 GEMM:  H[nRows,128] = A[nRows,128] @ W[128,128]
// 256 thr = 8 waves; wave w -> rows [blk*128 + w*16, +16), 8 16x16 N-tiles.
// W is staged into LDS in K-pair swizzled form: lW2[kp][n] = {W[2kp][n],
// W[2kp+1][n]} so each B fragment is ONE aligned ds_load_b64 into an even
// VGPR pair (no register shuffles). kp-row stride padded to 288 dwords so
// lanes 0-15 and 16-31 (kp vs kp+1) hit disjoint halves of the 64 LDS banks.
// A stream: unconditional global_load_b64 (clamped ptr + 0/1 mask), software
// pipelined one k-step ahead of the WMMA burst.
// V_WMMA_F32_16X16X4_F32, K stepped by 4 (32 steps, 8 tiles => 256 wmma/wave).
// A frag (16x4): lanes 0-15 hold K={k0,k0+1}, lanes 16-31 K={k0+2,k0+3}, row=lane&15
// B frag (4x16): mirrored K split, col = lane&15
// C/D (16x16):   vgpr v -> row v + 8*(lane>>4), col = lane&15
// ---------------------------------------------------------------------------
__global__ __launch_bounds__(256) void gemm_node_wmma(
    const float* __restrict__ A, const float* __restrict__ W,
    float* __restrict__ H, int nRows)
{
    __shared__ float lW2[KPAIRS * PSTRIDE];          // 72 KB of 320 KB WGP LDS

    // stage W as K-pairs: pair (kp, n) at dword offset kp*PSTRIDE + 2n
    for (int idx = threadIdx.x; idx < KPAIRS * DH; idx += 256) {
        const int kp = idx >> 7;          // 0..63
        const int n  = idx & (DH - 1);    // 0..127  (coalesced global reads)
        v2f w;
        w.x = W[(size_t)(2 * kp + 0) * DH + n];
        w.y = W[(size_t)(2 * kp + 1) * DH + n];
        *(v2f*)&lW2[kp * PSTRIDE + 2 * n] = w;       // ds_store_b64, bank-clean
    }
    __syncthreads();

    const int lane = threadIdx.x & 31;
    const int wave = threadIdx.x >> 5;
    const int rowBase = blockIdx.x * 128 + wave * 16;

    const int rlo = lane & 15;   // tile row (A) / tile col (B,D)
    const int hi  = lane >> 4;   // K-half select

    const int   arow    = rowBase + rlo;
    const float rowMask = (arow < nRows) ? 1.0f : 0.0f;     // no EXEC change
    const float* Arow   = A + (size_t)(arow < nRows ? arow : 0) * DH;

    v8f acc[8];
#pragma unroll
    for (int t = 0; t < 8; ++t) acc[t] = {};

    // software-pipelined A stream
    v2f araw = *(const v2f*)(Arow + 2 * hi);         // k0 = 0 fragment
    for (int k0 = 0; k0 < DH; k0 += 4) {
        v2f araw_next = araw;
        if (k0 + 4 < DH)
            araw_next = *(const v2f*)(Arow + (k0 + 4) + 2 * hi);  // prefetch

        v2f a;
        a.x = araw.x * rowMask;
        a.y = araw.y * rowMask;

        const int kp = (k0 >> 1) + hi;
        const float* bp = lW2 + (size_t)kp * PSTRIDE + 2 * rlo;
#pragma unroll
        for (int t = 0; t < 8; ++t) {
            const v2f b = *(const v2f*)(bp + 32 * t);   // single ds_load_b64
            acc[t] = __builtin_amdgcn_wmma_f32_16x16x4_f32(
                false, a, false, b, (short)0, acc[t], false, false);
        }
        araw = araw_next;
    }

#pragma unroll
    for (int t = 0; t < 8; ++t) {
        const int n0 = t * 16;
#pragma unroll
        for (int v = 0; v < 8; ++v) {
            const int r = rowBase + v + 8 * hi;
            if (r < nRows) H[(size_t)r * DH + n0 + rlo] = acc[t][v];
        }
    }
}

// ---------------------------------------------------------------------------
// Edge scatter: agg[dst] += h[src] * invs[src]*invs[dst]
// One wave32 per edge; each lane moves 4 features (float4 load, 4 f32 atomics).
// h table (25.6 MB) lives in the 192 MB L2 -> gathers/atomics are L2-resident.
// ---------------------------------------------------------------------------
__global__ __launch_bounds__(256) void scatter_edges(
    const float* __restrict__ h, const int* __restrict__ src,
    const int* __restrict__ dst, const float* __restrict__ invs,
    float* __restrict__ agg, int nEdges)
{
    const int gid  = blockIdx.x * blockDim.x + threadIdx.x;
    const int e    = gid >> 5;
    if (e >= nEdges) return;
    const int lane = gid & 31;

    const int s = src[e];
    const int d = dst[e];
    const float norm = invs[s] * invs[d];

    const float4 v = ((const float4*)(h + (size_t)s * DH))[lane];
    float* out = agg + (size_t)d * DH + lane * 4;
    atomicAdd(out + 0, v.x * norm);
    atomicAdd(out + 1, v.y * norm);
    atomicAdd(out + 2, v.z * norm);
    atomicAdd(out + 3, v.w * norm);
}

// ---------------------------------------------------------------------------
// Post-aggregate: agg += h * (1/deg) + bias ; ReLU in place; accumulate BN stats
// thread = feature f (t&127), rows strided; 1 atomic per thread at end.
// ---------------------------------------------------------------------------
__global__ __launch_bounds__(256) void post_relu_stats(
    const float* __restrict__ h, const float* __restrict__ invs,
    const float* __restrict__ bias, float* __restrict__ agg,
    float* __restrict__ sums, float* __restrict__ sumsq, int nNodes)
{
    const int f   = threadIdx.x & (DH - 1);
    const int sub = threadIdx.x >> 7;         // 0/1 -> 2 rows per block step
    const float bb = bias[f];

    float lsum = 0.0f, lsq = 0.0f;
    for (int i = blockIdx.x * 2 + sub; i < nNodes; i += gridDim.x * 2) {
        const float inv  = invs[i];
        const float invd = inv * inv;         // 1/deg
        const size_t off = (size_t)i * DH + f;
        float v = agg[off] + h[off] * invd + bb;
        v = v > 0.0f ? v : 0.0f;              // ReLU (feeds BN stats, per reference)
        agg[off] = v;
        lsum += v;
        lsq  += v * v;
    }
    atomicAdd(&sums[f],  lsum);
    atomicAdd(&sumsq[f], lsq);
}

// mean/var -> scale/shift (1 block, 128 threads); biased variance like reference
__global__ void bn_finalize(const float* __restrict__ sums,
                            const float* __restrict__ sumsq,
                            const float* __restrict__ g, const float* __restrict__ be,
                            float* __restrict__ scale, float* __restrict__ shift,
                            float invN)
{
    const int f = threadIdx.x;
    const float mean = sums[f] * invN;
    const float var  = sumsq[f] * invN - mean * mean;
    const float sc   = g[f] * rsqrtf(var + BN_EPS);
    scale[f] = sc;
    shift[f] = be[f] - mean * sc;
}

__global__ __launch_bounds__(256) void bn_apply(
    const float* __restrict__ in, const float* __restrict__ scale,
    const float* __restrict__ shift, float* __restrict__ out, size_t total)
{
    const size_t idx = (size_t)blockIdx.x * blockDim.x + threadIdx.x;
    if (idx >= total) return;
    const int f = (int)(idx & (DH - 1));
    out[idx] = in[idx] * scale[f] + shift[f];
}

// ---------------------------------------------------------------------------
// Global max pool per graph: order-preserving u32 keys + atomicMax
// ---------------------------------------------------------------------------
__device__ __forceinline__ unsigned f32_to_ordered(float x) {
    const unsigned u = __float_as_uint(x);
    return (u & 0x80000000u) ? ~u : (u | 0x80000000u);
}
__device__ __forceinline__ float ordered_to_f32(unsigned k) {
    const unsigned u = (k & 0x80000000u) ? (k & 0x7FFFFFFFu) : ~k;
    return __uint_as_float(u);
}

__global__ __launch_bounds__(256) void pool_max(
    const float* __restrict__ h, const int* __restrict__ batch,
    unsigned* __restrict__ keys, int nNodes)
{
    const size_t idx = (size_t)blockIdx.x * blockDim.x + threadIdx.x;
    const int i = (int)(idx >> 7);
    if (i >= nNodes) return;
    const int f = (int)(idx & (DH - 1));
    atomicMax(&keys[batch[i] * DH + f], f32_to_ordered(h[idx]));
}

// ---------------------------------------------------------------------------
// Predictor MLP: out[g] = relu(pooled[g]@Wp1 + bp1) @ Wp2 + bp2
// one block per graph; 256 threads = D_PRED
// ---------------------------------------------------------------------------
__global__ __launch_bounds__(256) void mlp_kernel(
    const unsigned* __restrict__ keys, const float* __restrict__ Wp1,
    const float* __restrict__ bp1, const float* __restrict__ Wp2,
    const float* __restrict__ bp2, float* __restrict__ out)
{
    __shared__ float pooled[DH];
    __shared__ float red[D_PRED];
    const int g = blockIdx.x;
    const int j = threadIdx.x;

    if (j < DH) pooled[j] = ordered_to_f32(keys[g * DH + j]);
    __syncthreads();

    float acc = bp1[j];
    for (int k = 0; k < DH; ++k) acc += pooled[k] * Wp1[k * D_PRED + j];
    acc = acc > 0.0f ? acc : 0.0f;
    red[j] = acc * Wp2[j];
    __syncthreads();

    for (int s = D_PRED / 2; s > 0; s >>= 1) {
        if (j < s) red[j] += red[j + s];
        __syncthreads();
    }
    if (j == 0) out[g] = red[0] + bp2[0];
}

// ---------------------------------------------------------------------------
// Host-side launcher
// ---------------------------------------------------------------------------
extern "C" void kernel_launch(void* const* d_in, const int* in_sizes, int n_in,
                              void* d_out, int out_size, void* d_ws, size_t ws_size,
                              hipStream_t stream)
{
    const float* x    = (const float*)d_in[0];
    const int*   ei   = (const int*)  d_in[1];
    const int*   bat  = (const int*)  d_in[2];
    const float* W1   = (const float*)d_in[3];
    const float* b1   = (const float*)d_in[4];
    const float* g1   = (const float*)d_in[5];
    const float* be1  = (const float*)d_in[6];
    const float* W2   = (const float*)d_in[7];
    const float* b2   = (const float*)d_in[8];
    const float* g2   = (const float*)d_in[9];
    const float* be2  = (const float*)d_in[10];
    const float* Wp1  = (const float*)d_in[11];
    const float* bp1  = (const float*)d_in[12];
    const float* Wp2  = (const float*)d_in[13];
    const float* bp2  = (const float*)d_in[14];
    float* out = (float*)d_out;

    const int* src = ei;
    const int* dst = ei + N_EDGES;

    // workspace layout (floats)
    const size_t NH = (size_t)N_NODES * DH;         // 6.4M
    float*    bufA  = (float*)d_ws;                 // node buffer A
    float*    bufB  = bufA + NH;                    // node buffer B
    float*    deg   = bufB + NH;                    // [N]
    float*    invs  = deg  + N_NODES;               // [N]
    float*    sums  = invs + N_NODES;               // [128]
    float*    sumsq = sums + DH;                    // [128]
    float*    scale = sumsq + DH;                   // [128]
    float*    shift = scale + DH;                   // [128]
    unsigned* keys  = (unsigned*)(shift + DH);      // [64*128]

    const int T = 256;
    const int gNH    = (int)((NH + T - 1) / T);
    const int gNode  = (N_NODES + T - 1) / T;
    const int gEdge  = (N_EDGES + T - 1) / T;
    const int gScat  = (int)(((size_t)N_EDGES * 32 + T - 1) / T);
    const int gGemm  = (N_NODES + 127) / 128;
    const int gPool  = (int)((NH + T - 1) / T);

    // ---- degree / norm -----------------------------------------------------
    zero_kernel<<<gNode, T, 0, stream>>>(deg, N_NODES);
    degree_kernel<<<gEdge, T, 0, stream>>>(dst, deg, N_EDGES);
    invsqrt_kernel<<<gNode, T, 0, stream>>>(deg, invs, N_NODES);

    // ---- layer 1: h1 = x@W1 ; aggregate ; relu+BN --------------------------
    gemm_node_wmma<<<gGemm, T, 0, stream>>>(x, W1, bufA, N_NODES);
    zero_kernel<<<gNH, T, 0, stream>>>(bufB, NH);
    zero_kernel<<<1, T, 0, stream>>>(sums, 2 * DH);           // sums+sumsq
    scatter_edges<<<gScat, T, 0, stream>>>(bufA, src, dst, invs, bufB, N_EDGES);
    post_relu_stats<<<256, T, 0, stream>>>(bufA, invs, b1, bufB, sums, sumsq, N_NODES);
    bn_finalize<<<1, DH, 0, stream>>>(sums, sumsq, g1, be1, scale, shift,
                                      1.0f / (float)N_NODES);
    bn_apply<<<gNH, T, 0, stream>>>(bufB, scale, shift, bufA, NH);

    // ---- layer 2: h2 = h1@W2 ; aggregate ; relu+BN -------------------------
    gemm_node_wmma<<<gGemm, T, 0, stream>>>(bufA, W2, bufB, N_NODES);
    zero_kernel<<<gNH, T, 0, stream>>>(bufA, NH);             // bufA becomes agg2
    zero_kernel<<<1, T, 0, stream>>>(sums, 2 * DH);
    scatter_edges<<<gScat, T, 0, stream>>>(bufB, src, dst, invs, bufA, N_EDGES);
    post_relu_stats<<<256, T, 0, stream>>>(bufB, invs, b2, bufA, sums, sumsq, N_NODES);
    bn_finalize<<<1, DH, 0, stream>>>(sums, sumsq, g2, be2, scale, shift,
                                      1.0f / (float)N_NODES);
    bn_apply<<<gNH, T, 0, stream>>>(bufA, scale, shift, bufB, NH);

    // ---- pool + MLP --------------------------------------------------------
    zero_kernel<<<(NUM_GRAPHS * DH + T - 1) / T, T, 0, stream>>>((float*)keys,
                                                                 NUM_GRAPHS * DH);
    pool_max<<<gPool, T, 0, stream>>>(bufB, bat, keys, N_NODES);
    mlp_kernel<<<NUM_GRAPHS, D_PRED, 0, stream>>>(keys, Wp1, bp1, Wp2, bp2, out);
}